// Model_29463475650789
// MI455X (gfx1250) — compile-verified
//
#include <hip/hip_runtime.h>
#include <hip/hip_bf16.h>

typedef __attribute__((ext_vector_type(2))) float v2f;
typedef __attribute__((ext_vector_type(8))) float v8f;

#define NS_N 100000
#define NP_N 20000
#define E_N  500000
#define FH   128

// ---------------- zero ----------------
__global__ void zero_f32(float* __restrict__ p, long n) {
    long i = (long)blockIdx.x * blockDim.x + threadIdx.x;
    if (i < n) p[i] = 0.0f;
}

// ---------------- gather x_play = playlist_embed[playlist_node_id] ----------------
__global__ void gather_rows(const float* __restrict__ src, const int* __restrict__ idx,
                            float* __restrict__ dst, int nrows) {
    long t = (long)blockIdx.x * blockDim.x + threadIdx.x;
    int row = (int)(t >> 5);
    int lane = (int)(t & 31);
    if (row < nrows) {
        const float4* s = (const float4*)(src + (size_t)idx[row] * FH);
        float4* d = (float4*)(dst + (size_t)row * FH);
        d[lane] = s[lane];
    }
}

// ---------------- edge degree counts ----------------
__global__ void count_edges(const int* __restrict__ e_song, const int* __restrict__ e_play,
                            float* __restrict__ cnt_s, float* __restrict__ cnt_p) {
    int e = blockIdx.x * blockDim.x + threadIdx.x;
    if (e < E_N) {
        atomicAdd(&cnt_s[e_song[e]], 1.0f);
        atomicAdd(&cnt_p[e_play[e]], 1.0f);
    }
}

// ---------------- scatter-sum of 128-wide rows: agg[e_dst] += src[e_src] ----------------
__global__ void scatter_sum(const float* __restrict__ src, const int* __restrict__ e_src,
                            const int* __restrict__ e_dst, float* __restrict__ agg) {
    long t = (long)blockIdx.x * blockDim.x + threadIdx.x;
    int e = (int)(t >> 5);
    int lane = (int)(t & 31);
    if (e < E_N) {
        int s = e_src[e];
        int d = e_dst[e];
        float4 v = ((const float4*)(src + (size_t)s * FH))[lane];
        float* a = agg + (size_t)d * FH + (size_t)lane * 4;
        atomicAdd(a + 0, v.x);
        atomicAdd(a + 1, v.y);
        atomicAdd(a + 2, v.z);
        atomicAdd(a + 3, v.w);
    }
}

// ---------------- mean normalize: agg[n,:] /= max(cnt[n],1) ----------------
__global__ void normalize_mean(float* __restrict__ agg, const float* __restrict__ cnt, int n) {
    long t = (long)blockIdx.x * blockDim.x + threadIdx.x;
    int row = (int)(t >> 5);
    int lane = (int)(t & 31);
    if (row < n) {
        float inv = 1.0f / fmaxf(cnt[row], 1.0f);
        float4* p = (float4*)(agg + (size_t)row * FH);
        float4 v = p[lane];
        v.x *= inv; v.y *= inv; v.z *= inv; v.w *= inv;
        p[lane] = v;
    }
}

// ---------------- pack weights into WMMA B-operand order ----------------
// Wp viewed as [64][128] of v2f:  Wp[p][col] = { W[2p][col], W[2p+1][col] }
// so a lane's B operand for (k, khalf, col) is one aligned 8B load at p = k*2 + khalf.
__global__ void pack_w(const float* __restrict__ W, float* __restrict__ Wp) {
    int t = blockIdx.x * blockDim.x + threadIdx.x;   // 64*128 = 8192 threads
    if (t < 64 * FH) {
        int p = t >> 7;
        int col = t & 127;
        v2f v;
        v.x = W[(size_t)(2 * p) * FH + col];
        v.y = W[(size_t)(2 * p + 1) * FH + col];
        ((v2f*)Wp)[(size_t)p * FH + col] = v;
    }
}

// ---------------- SAGE dense part via f32 WMMA ----------------
// out[n,:] = maybe_relu( Aagg @ Wl + bias + Xdst @ Wr )
// One wave per 16-row tile. K is the OUTER loop; all 8 N-tile accumulators stay
// live in registers (8 x v8f = 64 VGPRs, constant-indexed), A loaded per K step.
__global__ __launch_bounds__(256)
void sage_gemm(const float* __restrict__ Aagg, const float* __restrict__ Xdst,
               const float* __restrict__ Wlp, const float* __restrict__ Wrp,
               const float* __restrict__ bias, float* __restrict__ out,
               int nrows, int do_relu) {
    int waveId = (int)(((long)blockIdx.x * blockDim.x + threadIdx.x) >> 5); // uniform per wave
    int lane = threadIdx.x & 31;
    int ntiles = nrows >> 4;
    if (waveId >= ntiles) return;                  // wave-uniform exit; EXEC stays full below

    int m = lane & 15;          // row within tile (A) / column within N-tile (B, C/D)
    int khalf = lane >> 4;      // selects K pair within the K=4 step
    size_t rowBase = (size_t)(waveId * 16 + m) * FH;

    const v2f* wl = (const v2f*)Wlp;
    const v2f* wr = (const v2f*)Wrp;

    v8f acc[8];
    #pragma unroll
    for (int nt = 0; nt < 8; ++nt) acc[nt] = (v8f){};

    #pragma unroll 4
    for (int k = 0; k < 32; ++k) {
        int kk = k * 4 + khalf * 2;                    // even -> 8B aligned
        v2f a1 = *(const v2f*)(Aagg + rowBase + kk);
        v2f a2 = *(const v2f*)(Xdst + rowBase + kk);
        size_t pBase = (size_t)(k * 2 + khalf) * FH + m;   // v2f index; +nt*16 per tile
        #pragma unroll
        for (int nt = 0; nt < 8; ++nt) {
            v2f b1 = wl[pBase + nt * 16];
            acc[nt] = __builtin_amdgcn_wmma_f32_16x16x4_f32(
                false, a1, false, b1, (short)0, acc[nt], false, false);
        }
        #pragma unroll
        for (int nt = 0; nt < 8; ++nt) {
            v2f b2 = wr[pBase + nt * 16];
            acc[nt] = __builtin_amdgcn_wmma_f32_16x16x4_f32(
                false, a2, false, b2, (short)0, acc[nt], false, false);
        }
    }

    // Epilogue: bias add, optional relu, store.
    // C/D layout: VGPR r -> M=r (lanes 0-15) / M=r+8 (lanes 16-31), N = lane&15.
    #pragma unroll
    for (int nt = 0; nt < 8; ++nt) {
        int col = nt * 16 + m;
        float bcol = bias[col];
        #pragma unroll
        for (int r = 0; r < 8; ++r) {
            int row = waveId * 16 + r + (khalf << 3);
            float v = acc[nt][r] + bcol;
            if (do_relu) v = fmaxf(v, 0.0f);
            out[(size_t)row * FH + col] = v;
        }
    }
}

extern "C" void kernel_launch(void* const* d_in, const int* in_sizes, int n_in,
                              void* d_out, int out_size, void* d_ws, size_t ws_size,
                              hipStream_t stream) {
    const float* song_x = (const float*)d_in[0];
    const int*   pid    = (const int*)d_in[1];
    const int*   e_song = (const int*)d_in[2];
    const int*   e_play = (const int*)d_in[3];
    const float* pemb   = (const float*)d_in[4];
    const float* Wl1_sp = (const float*)d_in[5];
    const float* Wr1_sp = (const float*)d_in[6];
    const float* b1_sp  = (const float*)d_in[7];
    const float* Wl1_ps = (const float*)d_in[8];
    const float* Wr1_ps = (const float*)d_in[9];
    const float* b1_ps  = (const float*)d_in[10];
    const float* Wl2_sp = (const float*)d_in[11];
    const float* Wr2_sp = (const float*)d_in[12];
    const float* b2_sp  = (const float*)d_in[13];
    const float* Wl2_ps = (const float*)d_in[14];
    const float* Wr2_ps = (const float*)d_in[15];
    const float* b2_ps  = (const float*)d_in[16];

    // workspace layout (floats)
    float* ws    = (float*)d_ws;
    float* agg_p = ws;                                  // NP*128
    float* agg_s = agg_p + (size_t)NP_N * FH;           // NS*128
    float* cnt_p = agg_s + (size_t)NS_N * FH;           // NP
    float* cnt_s = cnt_p + NP_N;                        // NS
    float* xplay = cnt_s + NS_N;                        // NP*128
    float* p1    = xplay + (size_t)NP_N * FH;           // NP*128
    float* s1    = p1 + (size_t)NP_N * FH;              // NS*128
    float* wpk   = s1 + (size_t)NS_N * FH;              // 8 * 64*128*2 = 131072 floats
    float* Wl1_sp_p = wpk + 0 * 16384;
    float* Wr1_sp_p = wpk + 1 * 16384;
    float* Wl1_ps_p = wpk + 2 * 16384;
    float* Wr1_ps_p = wpk + 3 * 16384;
    float* Wl2_sp_p = wpk + 4 * 16384;
    float* Wr2_sp_p = wpk + 5 * 16384;
    float* Wl2_ps_p = wpk + 6 * 16384;
    float* Wr2_ps_p = wpk + 7 * 16384;

    float* s2_out = (float*)d_out;                      // [NS,128] first
    float* p2_out = s2_out + (size_t)NS_N * FH;         // then [NP,128]

    const int T = 256;

    // pack all 8 weight matrices
    pack_w<<<32, T, 0, stream>>>(Wl1_sp, Wl1_sp_p);
    pack_w<<<32, T, 0, stream>>>(Wr1_sp, Wr1_sp_p);
    pack_w<<<32, T, 0, stream>>>(Wl1_ps, Wl1_ps_p);
    pack_w<<<32, T, 0, stream>>>(Wr1_ps, Wr1_ps_p);
    pack_w<<<32, T, 0, stream>>>(Wl2_sp, Wl2_sp_p);
    pack_w<<<32, T, 0, stream>>>(Wr2_sp, Wr2_sp_p);
    pack_w<<<32, T, 0, stream>>>(Wl2_ps, Wl2_ps_p);
    pack_w<<<32, T, 0, stream>>>(Wr2_ps, Wr2_ps_p);

    long nZero1 = (long)NP_N * FH + (long)NS_N * FH + NP_N + NS_N; // agg_p..cnt_s contiguous
    zero_f32<<<(int)((nZero1 + T - 1) / T), T, 0, stream>>>(agg_p, nZero1);

    gather_rows<<<(NP_N * 32 + T - 1) / T, T, 0, stream>>>(pemb, pid, xplay, NP_N);
    count_edges<<<(E_N + T - 1) / T, T, 0, stream>>>(e_song, e_play, cnt_s, cnt_p);

    // ---- layer 1 aggregation ----
    long nScat = (long)E_N * 32;
    scatter_sum<<<(int)((nScat + T - 1) / T), T, 0, stream>>>(song_x, e_song, e_play, agg_p);
    scatter_sum<<<(int)((nScat + T - 1) / T), T, 0, stream>>>(xplay, e_play, e_song, agg_s);
    normalize_mean<<<(NP_N * 32 + T - 1) / T, T, 0, stream>>>(agg_p, cnt_p, NP_N);
    normalize_mean<<<(NS_N * 32 + T - 1) / T, T, 0, stream>>>(agg_s, cnt_s, NS_N);

    // ---- layer 1 dense (WMMA) ----
    int tilesP = NP_N / 16, tilesS = NS_N / 16; // 8 waves per 256-thread block
    sage_gemm<<<(tilesP + 7) / 8, T, 0, stream>>>(agg_p, xplay, Wl1_sp_p, Wr1_sp_p, b1_sp, p1, NP_N, 1);
    sage_gemm<<<(tilesS + 7) / 8, T, 0, stream>>>(agg_s, song_x, Wl1_ps_p, Wr1_ps_p, b1_ps, s1, NS_N, 1);

    // ---- layer 2 aggregation (re-zero agg buffers; counts unchanged) ----
    long nZero2 = (long)NP_N * FH + (long)NS_N * FH;
    zero_f32<<<(int)((nZero2 + T - 1) / T), T, 0, stream>>>(agg_p, nZero2);
    scatter_sum<<<(int)((nScat + T - 1) / T), T, 0, stream>>>(s1, e_song, e_play, agg_p);
    scatter_sum<<<(int)((nScat + T - 1) / T), T, 0, stream>>>(p1, e_play, e_song, agg_s);
    normalize_mean<<<(NP_N * 32 + T - 1) / T, T, 0, stream>>>(agg_p, cnt_p, NP_N);
    normalize_mean<<<(NS_N * 32 + T - 1) / T, T, 0, stream>>>(agg_s, cnt_s, NS_N);

    // ---- layer 2 dense (WMMA), no relu, straight to output ----
    sage_gemm<<<(tilesP + 7) / 8, T, 0, stream>>>(agg_p, p1, Wl2_sp_p, Wr2_sp_p, b2_sp, p2_out, NP_N, 0);
    sage_gemm<<<(tilesS + 7) / 8, T, 0, stream>>>(agg_s, s1, Wl2_ps_p, Wr2_ps_p, b2_ps, s2_out, NS_N, 0);
}